// ScaledDotProductAttention_8220567404860
// MI455X (gfx1250) — compile-verified
//
#include <hip/hip_runtime.h>
#include <math.h>

typedef __attribute__((ext_vector_type(2))) float        v2f;
typedef __attribute__((ext_vector_type(8))) float        v8f;
typedef __attribute__((ext_vector_type(4))) unsigned int v4u;
typedef __attribute__((ext_vector_type(4))) int          v4i;
typedef __attribute__((ext_vector_type(8))) int          v8i;

#define B_   4
#define H_   16
#define LQ_  1024
#define LK_  1024
#define DH_  64
#define QB   32                    // query rows per workgroup
#define SS_STRIDE (LK_ + 4)        // padded score-row stride
#define SQ_STRIDE (DH_ + 4)        // padded Q-row stride
#define KSLAB_STRIDE 132           // 128 dwords + TDM pad (interval=6, amount=3)
#define VSLAB_STRIDE 68            // 64 dwords + TDM pad (interval=5, amount=3)
#define SLAB_FLOATS  (128 * VSLAB_STRIDE)   // 8704 floats: max(64*132, 128*68)

// ---- Tensor Data Mover: 2-D tile load into LDS (D# per CDNA5 ISA ch.8) ----
__device__ __forceinline__ void tdm_load_2d(unsigned lds_byte_off,
                                            const void* gaddr,
                                            unsigned tensor_d0, unsigned tensor_d1,
                                            unsigned tile_d0,   unsigned tile_d1,
                                            unsigned stride0_elems,
                                            unsigned pad_interval, unsigned pad_amount) {
  const unsigned long long ga = (unsigned long long)gaddr;
  v4u g0;
  g0[0] = 1u;                                             // count=1 (valid user D#)
  g0[1] = lds_byte_off;                                   // lds_addr
  g0[2] = (unsigned)(ga & 0xFFFFFFFFu);                   // global_addr[31:0]
  g0[3] = (unsigned)(((ga >> 32) & 0x01FFFFFFu) | (2u << 30)); // ga[56:32] | type=2

  v8i g1;
  g1[0] = (int)((2u << 16)                                // data_size = 4 bytes
              | (1u << 20)                                // pad_enable
              | (pad_interval << 22)
              | (pad_amount   << 25));
  g1[1] = (int)((tensor_d0 & 0xFFFFu) << 16);             // tensor_dim0[15:0]
  g1[2] = (int)(((tensor_d0 >> 16) & 0xFFFFu)             // tensor_dim0[31:16]
              | ((tensor_d1 & 0xFFFFu) << 16));           // tensor_dim1[15:0]
  g1[3] = (int)(((tensor_d1 >> 16) & 0xFFFFu)             // tensor_dim1[31:16]
              | ((tile_d0 & 0xFFFFu) << 16));             // tile_dim0
  g1[4] = (int)(tile_d1 & 0xFFFFu);                       // tile_dim1 (tile_dim2=0)
  g1[5] = (int)stride0_elems;                             // tensor_dim0_stride[31:0]
  g1[6] = 0;                                              // stride0[47:32], stride1 lo
  g1[7] = 0;

  const v4i gz4 = {0, 0, 0, 0};                           // 2-D: groups 2/3 unused
  const v8i gz8 = {0, 0, 0, 0, 0, 0, 0, 0};
  __builtin_amdgcn_tensor_load_to_lds(g0, g1, gz4, gz4, gz8, 0);
}

__global__ __launch_bounds__(256)
void fused_attn_f32_wmma(const float* __restrict__ qg,
                         const float* __restrict__ kg,
                         const float* __restrict__ vg,
                         const float* __restrict__ pre,
                         const float* __restrict__ amask,         // [1,LQ,LK]
                         const unsigned char* __restrict__ kmask, // [B,LK] bool
                         float* __restrict__ out,                 // [B,H,LQ,DH]
                         float* __restrict__ out_w) {             // [B,H,LQ,LK]
  extern __shared__ float smem[];
  float* sS    = smem;                               // [QB][SS_STRIDE]
  float* sQ    = smem + QB * SS_STRIDE;              // [QB][SQ_STRIDE]
  float* sSlab = sQ   + QB * SQ_STRIDE;              // K slab 64x132 / V slab 128x68
  const unsigned slab_off = (unsigned)(unsigned long long)(const void*)sSlab; // LDS byte addr

  const int nqb = LQ_ / QB;
  const int qb  = blockIdx.x % nqb;
  const int bh  = blockIdx.x / nqb;
  const int b   = bh / H_;
  const int q0  = qb * QB;

  const float* Qb = qg  + (size_t)bh * LQ_ * DH_;
  const float* Kb = kg  + (size_t)bh * DH_ * LK_;    // [DH, LK] = K^T
  const float* Vb = vg  + (size_t)bh * LK_ * DH_;
  const float* Pb = pre + (size_t)bh * LQ_ * LK_;
  const unsigned char* Mb = kmask + (size_t)b * LK_;

  const int tid  = threadIdx.x;
  const int wave = tid >> 5;
  const int lane = tid & 31;
  const int lh   = lane >> 4;
  const int ll   = lane & 15;

  // ---------- Phase 0: stage Q block (32 x 64) into LDS ----------
  for (int i = tid; i < QB * DH_; i += 256) {
    const int r = i >> 6, c = i & 63;
    sQ[r * SQ_STRIDE + c] = Qb[(size_t)(q0 + r) * DH_ + c];
  }

  const float scale   = 0.125f;                      // 64^-0.5
  const float neg_inf = -__builtin_inff();

  // ---------- Phase 1: S = Q@K^T * scale + pre + amask, key-mask -> -inf ----------
  // 8 slabs of 128 key columns; slab DMA'd into LDS by the TDM.
  for (int s = 0; s < 8; ++s) {
    if (tid == 0) {
      // K^T tile: 64 rows (dh) x 128 cols, row stride LK; pad -> LDS stride 132
      tdm_load_2d(slab_off, Kb + (size_t)s * 128,
                  /*tensor*/ 128, 64, /*tile*/ 128, 64,
                  /*stride0*/ LK_, /*pad_interval*/ 6, /*pad_amount*/ 3);
      __builtin_amdgcn_s_wait_tensorcnt(0);
    }
    __syncthreads();                                 // slab (and phase-0 Q) visible

    const int cl = wave * 16 + ll;                   // column within slab (per wave tile)
    const int n  = s * 128 + cl;                     // global key column
    v8f acc0 = {}, acc1 = {};
    #pragma unroll
    for (int kk = 0; kk < 16; ++kk) {
      const int ka = kk * 4 + lh * 2;
      v2f bf, a0, a1;
      bf.x = sSlab[(ka)     * KSLAB_STRIDE + cl];
      bf.y = sSlab[(ka + 1) * KSLAB_STRIDE + cl];
      a0.x = sQ[(ll)      * SQ_STRIDE + ka];
      a0.y = sQ[(ll)      * SQ_STRIDE + ka + 1];
      a1.x = sQ[(16 + ll) * SQ_STRIDE + ka];
      a1.y = sQ[(16 + ll) * SQ_STRIDE + ka + 1];
      acc0 = __builtin_amdgcn_wmma_f32_16x16x4_f32(false, a0, false, bf,
                                                   (short)0, acc0, false, false);
      acc1 = __builtin_amdgcn_wmma_f32_16x16x4_f32(false, a1, false, bf,
                                                   (short)0, acc1, false, false);
    }

    const bool masked = (Mb[n] != 0);
    #pragma unroll
    for (int i = 0; i < 8; ++i) {
      const int m0 = lh * 8 + i;                     // rows 0..15  (tile 0)
      const int m1 = 16 + m0;                        // rows 16..31 (tile 1)
      float s0 = acc0[i] * scale + Pb[(size_t)(q0 + m0) * LK_ + n]
                                 + amask[(size_t)(q0 + m0) * LK_ + n];
      float s1 = acc1[i] * scale + Pb[(size_t)(q0 + m1) * LK_ + n]
                                 + amask[(size_t)(q0 + m1) * LK_ + n];
      sS[m0 * SS_STRIDE + n] = masked ? neg_inf : s0;
      sS[m1 * SS_STRIDE + n] = masked ? neg_inf : s1;
    }
    __syncthreads();                                 // done reading slab
  }

  // ---------- Phase 2: row-wise softmax; write weights (global + LDS) ----------
  float* Wb = out_w + (size_t)bh * LQ_ * LK_;
  for (int rr = 0; rr < 4; ++rr) {
    const int r = wave * 4 + rr;
    float vals[32];
    float mx = neg_inf;
    #pragma unroll
    for (int j = 0; j < 32; ++j) {
      vals[j] = sS[r * SS_STRIDE + (j * 32 + lane)];
      mx = fmaxf(mx, vals[j]);
    }
    #pragma unroll
    for (int off = 16; off > 0; off >>= 1)
      mx = fmaxf(mx, __shfl_xor(mx, off, 32));

    float sum = 0.0f;
    #pragma unroll
    for (int j = 0; j < 32; ++j) {
      vals[j] = __expf(vals[j] - mx);
      sum += vals[j];
    }
    #pragma unroll
    for (int off = 16; off > 0; off >>= 1)
      sum += __shfl_xor(sum, off, 32);

    const float inv = 1.0f / sum;
    #pragma unroll
    for (int j = 0; j < 32; ++j) {
      const float w = vals[j] * inv;
      sS[r * SS_STRIDE + (j * 32 + lane)] = w;
      Wb[(size_t)(q0 + r) * LK_ + (j * 32 + lane)] = w;
    }
  }
  __syncthreads();

  // ---------- Phase 3: O = P @ V, V slabs DMA'd by the TDM ----------
  {
    const int tm = wave >> 2;                        // 0..1
    const int tn = wave & 3;                         // 0..3
    const int d  = tn * 16 + ll;

    v8f acc = {};
    for (int vs = 0; vs < 8; ++vs) {
      if (tid == 0) {
        // V tile: 128 rows (k) x 64 cols (dh), row stride DH; pad -> LDS stride 68
        tdm_load_2d(slab_off, Vb + (size_t)vs * 128 * DH_,
                    /*tensor*/ 64, 128, /*tile*/ 64, 128,
                    /*stride0*/ DH_, /*pad_interval*/ 5, /*pad_amount*/ 3);
        __builtin_amdgcn_s_wait_tensorcnt(0);
      }
      __syncthreads();                               // slab visible

      #pragma unroll
      for (int kk = 0; kk < 32; ++kk) {
        const int ka = kk * 4 + lh * 2;
        v2f a, bf;
        a.x  = sS[(tm * 16 + ll) * SS_STRIDE + vs * 128 + ka];
        a.y  = sS[(tm * 16 + ll) * SS_STRIDE + vs * 128 + ka + 1];
        bf.x = sSlab[(ka)     * VSLAB_STRIDE + d];
        bf.y = sSlab[(ka + 1) * VSLAB_STRIDE + d];
        acc = __builtin_amdgcn_wmma_f32_16x16x4_f32(false, a, false, bf,
                                                    (short)0, acc, false, false);
      }
      __syncthreads();                               // done reading slab
    }

    float* Ob = out + (size_t)bh * LQ_ * DH_;
    #pragma unroll
    for (int i = 0; i < 8; ++i) {
      const int m = tm * 16 + lh * 8 + i;
      Ob[(size_t)(q0 + m) * DH_ + d] = acc[i];
    }
  }
}

extern "C" void kernel_launch(void* const* d_in, const int* in_sizes, int n_in,
                              void* d_out, int out_size, void* d_ws, size_t ws_size,
                              hipStream_t stream) {
  (void)in_sizes; (void)n_in; (void)out_size; (void)d_ws; (void)ws_size;

  const float*         q   = (const float*)d_in[0];
  const float*         k   = (const float*)d_in[1];
  const float*         v   = (const float*)d_in[2];
  const float*         pre = (const float*)d_in[3];
  const float*         am  = (const float*)d_in[4];
  const unsigned char* km  = (const unsigned char*)d_in[5];

  float* out   = (float*)d_out;
  float* out_w = out + (size_t)B_ * H_ * LQ_ * DH_;

  const size_t lds_bytes =
      (size_t)(QB * SS_STRIDE + QB * SQ_STRIDE + SLAB_FLOATS) * sizeof(float);
  const dim3 grid(B_ * H_ * (LQ_ / QB));
  fused_attn_f32_wmma<<<grid, 256, lds_bytes, stream>>>(q, k, v, pre, am, km, out, out_w);
}